// HomoGNN_GraphConv_9715216023821
// MI455X (gfx1250) — compile-verified
//
#include <hip/hip_runtime.h>

typedef float v2f __attribute__((ext_vector_type(2)));
typedef float v8f __attribute__((ext_vector_type(8)));

#define HDIM 64
#define LDT 65  // padded LDS row stride (floats) -> conflict-free column access

// ---------------------------------------------------------------------------
// D[16x64] += A[16xK] @ W[Kx64], all fp32, via V_WMMA_F32_16X16X4_F32.
// A: row-major with stride lda (global or LDS). W: row-major [K,64].
// acc[n] covers output columns n*16..n*16+15.
// ---------------------------------------------------------------------------
__device__ __forceinline__ void wmma_f32_gemm(v8f acc[4], const float* __restrict__ A, int lda,
                                              const float* __restrict__ W, int K) {
  const int lane = threadIdx.x & 31;
  const int m    = lane & 15;          // M index (A) / N index (B)
  const int kh   = (lane >> 4) << 1;   // lanes 16..31 carry K+2, K+3
  for (int k = 0; k < K; k += 4) {
    v2f a;
    a.x = A[m * lda + k + kh];
    a.y = A[m * lda + k + kh + 1];
#pragma unroll
    for (int n = 0; n < 4; ++n) {
      v2f b;
      b.x = W[(k + kh) * HDIM + n * 16 + m];
      b.y = W[(k + kh + 1) * HDIM + n * 16 + m];
      acc[n] = __builtin_amdgcn_wmma_f32_16x16x4_f32(false, a, false, b, (short)0, acc[n],
                                                     false, false);
    }
  }
}

__device__ __forceinline__ void zero_acc(v8f acc[4]) {
  const v8f z = {0.f, 0.f, 0.f, 0.f, 0.f, 0.f, 0.f, 0.f};
#pragma unroll
  for (int n = 0; n < 4; ++n) acc[n] = z;
}

// Store C/D fragment (+bias, optional relu) to row-major dst with stride ldd.
__device__ __forceinline__ void store_tile(const v8f acc[4], float* dst, int ldd,
                                           const float* __restrict__ bias, bool relu) {
  const int lane = threadIdx.x & 31;
  const int col  = lane & 15;
  const int rb   = (lane >> 4) << 3;  // rows 0..7 or 8..15
#pragma unroll
  for (int n = 0; n < 4; ++n) {
    const float bv = bias ? bias[n * 16 + col] : 0.0f;
#pragma unroll
    for (int i = 0; i < 8; ++i) {
      float v = acc[n][i] + bv;
      if (relu) v = fmaxf(v, 0.0f);
      dst[(rb + i) * ldd + n * 16 + col] = v;
    }
  }
}

__device__ __forceinline__ void lds_fence() {
  asm volatile("s_wait_dscnt 0" ::: "memory");
}

// ---------------------------------------------------------------------------
// h = relu(x @ W1 + b1) @ W2 + b2        x:[N,128]  h:[N,64]
// one wave per 16-row tile, 8 waves per block
// ---------------------------------------------------------------------------
__global__ void embed_kernel(const float* __restrict__ x, const float* __restrict__ w1,
                             const float* __restrict__ b1, const float* __restrict__ w2,
                             const float* __restrict__ b2, float* __restrict__ h, int nTiles) {
  __shared__ float lds[8][16 * LDT];
  const int wave = threadIdx.x >> 5;
  const int tile = blockIdx.x * 8 + wave;
  if (tile >= nTiles) return;  // wave-uniform: EXEC stays all-ones
  const int row0 = tile * 16;
  float* tA = lds[wave];

  v8f acc[4];
  zero_acc(acc);
  wmma_f32_gemm(acc, x + (size_t)row0 * 128, 128, w1, 128);
  store_tile(acc, tA, LDT, b1, true);   // mid = relu(x@W1 + b1)
  lds_fence();

  zero_acc(acc);
  wmma_f32_gemm(acc, tA, LDT, w2, 64);
  store_tile(acc, h + (size_t)row0 * HDIM, HDIM, b2, false);  // h = mid@W2 + b2
}

// ---------------------------------------------------------------------------
// Fused GraphConv block for one layer:
//   t = (agg*invdeg) @ rel_w + rel_b + h @ root_w
//   u = relu(t @ pw1 + pb1)
//   h = relu(u @ pw2 + pb2)          (in-place on h: each wave owns its rows)
// ---------------------------------------------------------------------------
__global__ void conv_kernel(const float* __restrict__ agg, const float* __restrict__ invdeg,
                            float* __restrict__ h,
                            const float* __restrict__ rel_w, const float* __restrict__ rel_b,
                            const float* __restrict__ root_w,
                            const float* __restrict__ pw1, const float* __restrict__ pb1,
                            const float* __restrict__ pw2, const float* __restrict__ pb2,
                            int nTiles) {
  __shared__ float lds[8][2][16 * LDT];
  const int lane = threadIdx.x & 31;
  const int wave = threadIdx.x >> 5;
  const int tile = blockIdx.x * 8 + wave;
  if (tile >= nTiles) return;
  const int row0 = tile * 16;
  float* tA = lds[wave][0];
  float* tB = lds[wave][1];

  // stage scaled aggregation and current h into LDS
  for (int idx = lane; idx < 16 * HDIM; idx += 32) {
    const int r = idx >> 6, c = idx & 63;
    tA[r * LDT + c] = agg[(size_t)(row0 + r) * HDIM + c] * invdeg[row0 + r];
    tB[r * LDT + c] = h[(size_t)(row0 + r) * HDIM + c];
  }
  lds_fence();

  v8f acc[4];
  zero_acc(acc);
  wmma_f32_gemm(acc, tA, LDT, rel_w, 64);   // agg_scaled @ rel_w
  wmma_f32_gemm(acc, tB, LDT, root_w, 64);  // + h @ root_w
  store_tile(acc, tA, LDT, rel_b, false);   // t  (overwrite tA; GEMMs done reading it)
  lds_fence();

  zero_acc(acc);
  wmma_f32_gemm(acc, tA, LDT, pw1, 64);
  store_tile(acc, tB, LDT, pb1, true);      // u = relu(t@pw1 + pb1)
  lds_fence();

  zero_acc(acc);
  wmma_f32_gemm(acc, tB, LDT, pw2, 64);
  store_tile(acc, h + (size_t)row0 * HDIM, HDIM, pb2, true);  // h = relu(u@pw2 + pb2)
}

// ---------------------------------------------------------------------------
// sparse / reduction kernels
// ---------------------------------------------------------------------------
__global__ void zero_kernel(float* __restrict__ p, long n) {
  const long i = (long)blockIdx.x * blockDim.x + threadIdx.x;
  if (i < n) p[i] = 0.0f;
}

__global__ void deg_kernel(const int* __restrict__ dst, float* __restrict__ deg, int E) {
  const int e = blockIdx.x * blockDim.x + threadIdx.x;
  if (e < E) atomicAdd(&deg[dst[e]], 1.0f);
}

__global__ void invdeg_kernel(float* __restrict__ deg, int N) {
  const int i = blockIdx.x * blockDim.x + threadIdx.x;
  if (i < N) deg[i] = 1.0f / fmaxf(deg[i], 1.0f);
}

// agg[dst] += h[src]; one thread per (edge, 4-float chunk): 16 threads/edge
__global__ void scatter_kernel(const float* __restrict__ h, const int* __restrict__ src,
                               const int* __restrict__ dst, float* __restrict__ agg, int E) {
  const long gid = (long)blockIdx.x * blockDim.x + threadIdx.x;
  const int e = (int)(gid >> 4);
  if (e >= E) return;
  const int f = (int)(gid & 15) << 2;
  const int s = src[e], d = dst[e];
  const float4 v = *(const float4*)(h + (size_t)s * HDIM + f);
  float* o = agg + (size_t)d * HDIM + f;
  atomicAdd(o + 0, v.x);
  atomicAdd(o + 1, v.y);
  atomicAdd(o + 2, v.z);
  atomicAdd(o + 3, v.w);
}

__global__ void pool_kernel(const float* __restrict__ h, const int* __restrict__ batch,
                            float* __restrict__ pooled, float* __restrict__ cnt, int N) {
  const long gid = (long)blockIdx.x * blockDim.x + threadIdx.x;
  const int n = (int)(gid >> 4);
  if (n >= N) return;
  const int f = (int)(gid & 15) << 2;
  const int g = batch[n];
  const float4 v = *(const float4*)(h + (size_t)n * HDIM + f);
  float* o = pooled + (size_t)g * HDIM + f;
  atomicAdd(o + 0, v.x);
  atomicAdd(o + 1, v.y);
  atomicAdd(o + 2, v.z);
  atomicAdd(o + 3, v.w);
  if (f == 0) atomicAdd(&cnt[g], 1.0f);
}

__global__ void cls_kernel(const float* __restrict__ pooled, const float* __restrict__ cnt,
                           const float* __restrict__ w, const float* __restrict__ b,
                           float* __restrict__ out, int G, int C) {
  const int i = blockIdx.x * blockDim.x + threadIdx.x;
  if (i >= G * C) return;
  const int g = i / C, c = i % C;
  float s = 0.0f;
  for (int f = 0; f < HDIM; ++f) s += pooled[g * HDIM + f] * w[f * C + c];
  out[i] = s / fmaxf(cnt[g], 1.0f) + b[c];
}

// ---------------------------------------------------------------------------
extern "C" void kernel_launch(void* const* d_in, const int* in_sizes, int n_in,
                              void* d_out, int out_size, void* d_ws, size_t ws_size,
                              hipStream_t stream) {
  const float* x      = (const float*)d_in[0];
  const int*   ei     = (const int*)d_in[1];
  const int*   batch  = (const int*)d_in[2];
  const float* emb_w1 = (const float*)d_in[3];
  const float* emb_b1 = (const float*)d_in[4];
  const float* emb_w2 = (const float*)d_in[5];
  const float* emb_b2 = (const float*)d_in[6];
  const float* rel_w  = (const float*)d_in[7];
  const float* rel_b  = (const float*)d_in[8];
  const float* root_w = (const float*)d_in[9];
  const float* pw1    = (const float*)d_in[10];
  const float* pb1    = (const float*)d_in[11];
  const float* pw2    = (const float*)d_in[12];
  const float* pb2    = (const float*)d_in[13];
  const float* cls_w  = (const float*)d_in[14];
  const float* cls_b  = (const float*)d_in[15];

  const int N = in_sizes[2];                       // 50000
  const int E = in_sizes[1] / 2;                   // 1.25M
  const int C = in_sizes[15];                      // 10
  const int G = out_size / C;                      // 128
  const int L = in_sizes[7] / (HDIM * HDIM);       // 2

  const int* src = ei;
  const int* dst = ei + E;

  char* ws = (char*)d_ws;
  float* h      = (float*)ws; ws += (size_t)N * HDIM * sizeof(float);
  float* agg    = (float*)ws; ws += (size_t)N * HDIM * sizeof(float);
  float* deg    = (float*)ws; ws += (size_t)N * sizeof(float);
  float* pooled = (float*)ws; ws += (size_t)G * HDIM * sizeof(float);
  float* cnt    = (float*)ws;

  const int nTiles     = (N + 15) / 16;
  const int convBlocks = (nTiles + 7) / 8;

  // degrees
  zero_kernel<<<(N + 255) / 256, 256, 0, stream>>>(deg, N);
  deg_kernel<<<(E + 255) / 256, 256, 0, stream>>>(dst, deg, E);
  invdeg_kernel<<<(N + 255) / 256, 256, 0, stream>>>(deg, N);

  // node embedding MLP
  embed_kernel<<<convBlocks, 256, 0, stream>>>(x, emb_w1, emb_b1, emb_w2, emb_b2, h, nTiles);

  // conv layers
  for (int l = 0; l < L; ++l) {
    zero_kernel<<<(int)(((long)N * HDIM + 255) / 256), 256, 0, stream>>>(agg, (long)N * HDIM);
    scatter_kernel<<<(int)(((long)E * 16 + 255) / 256), 256, 0, stream>>>(h, src, dst, agg, E);
    conv_kernel<<<convBlocks, 256, 0, stream>>>(agg, deg, h,
                                                rel_w + (size_t)l * HDIM * HDIM,
                                                rel_b + (size_t)l * HDIM,
                                                root_w + (size_t)l * HDIM * HDIM,
                                                pw1 + (size_t)l * HDIM * HDIM,
                                                pb1 + (size_t)l * HDIM,
                                                pw2 + (size_t)l * HDIM * HDIM,
                                                pb2 + (size_t)l * HDIM, nTiles);
  }

  // mean pool + classifier
  zero_kernel<<<(G * HDIM + 255) / 256, 256, 0, stream>>>(pooled, (long)G * HDIM);
  zero_kernel<<<1, 256, 0, stream>>>(cnt, G);
  pool_kernel<<<(int)(((long)N * 16 + 255) / 256), 256, 0, stream>>>(h, batch, pooled, cnt, N);
  cls_kernel<<<(G * C + 255) / 256, 256, 0, stream>>>(pooled, cnt, cls_w, cls_b,
                                                      (float*)d_out, G, C);
}